// HydroSSMCell_71399536329341
// MI455X (gfx1250) — compile-verified
//
#include <hip/hip_runtime.h>
#include <math.h>

// Problem constants (match reference setup_inputs)
#define HIDDEN   128
#define U_DIM    8
#define Y_DIM    9
#define KDIM     17          // fan_in = U_DIM + Y_DIM
#define KPAD     20          // padded K (multiple of 4 for 16x16x4 WMMA)
#define B_SZ     64
#define L_SZ     4096
#define CHUNK    128         // timesteps per chunk (parallel-scan block)
#define NCHUNK   (L_SZ / CHUNK)   // 32
#define TILE     16          // timesteps per WMMA tile
#define NTILES   (CHUNK / TILE)   // 8

// 2 * log2(e): tanh(x) = 1 - 2/(exp2(x*C)+1)
#define TANH_C   2.885390081777927f

typedef float v2f __attribute__((ext_vector_type(2)));
typedef float v8f __attribute__((ext_vector_type(8)));

__device__ __forceinline__ float fast_sigmoid(float x) {
    return __builtin_amdgcn_rcpf(1.0f + __expf(-x));
}

// tanh(acc + bn) with the bias pre-folded: bnC = bn * TANH_C.
// One fma -> v_exp_f32 -> add -> v_rcp_f32 -> fma. Branch-free, saturating.
__device__ __forceinline__ float fused_bias_tanh(float acc, float bnC) {
    float t = fmaf(acc, TANH_C, bnC);
    float e = __builtin_amdgcn_exp2f(t);
    float r = __builtin_amdgcn_rcpf(e + 1.0f);
    return fmaf(-2.0f, r, 1.0f);
}

// ---------------------------------------------------------------------------
// Chunk kernel: computes z = tanh([u,y]·W + b) for its chunk via fp32 WMMA,
// then scans h_k = lam*h_{k-1} + (1-lam)*z_k across the chunk.
//   PHASE2 == false : start from h=0, emit only chunk-final state (workspace)
//   PHASE2 == true  : start from exact h_init (workspace), emit h_seq/h_last
// Block = 128 threads = 4 wave32; wave w owns hidden columns [32w, 32w+32).
// ---------------------------------------------------------------------------
template <bool PHASE2>
__global__ __launch_bounds__(128)
void hydro_chunk_kernel(const float* __restrict__ u,
                        const float* __restrict__ y,
                        const float* __restrict__ W,      // (17, 128) row-major
                        const float* __restrict__ bias,   // (128,)
                        const float* __restrict__ lam_p,  // (128,)
                        const float* __restrict__ hinit,  // (B, NCHUNK, H) [phase2]
                        float* __restrict__ local_out,    // (B, NCHUNK, H) [phase1]
                        float* __restrict__ h_seq,        // (B, L, H)      [phase2]
                        float* __restrict__ h_last)       // (B, H)         [phase2]
{
    __shared__ float s_inp[TILE][KPAD];    // input tile, cols 17..19 = 0
    __shared__ float s_z[TILE][HIDDEN];    // z tile (time-major)

    const int b    = blockIdx.y;
    const int c    = blockIdx.x;
    const int tid  = threadIdx.x;
    const int wave = tid >> 5;
    const int lane = tid & 31;
    const int m16  = lane & 15;            // row index inside 16-wide dim
    const int hi   = lane >> 4;            // half-wave select
    const int k0   = hi * 2;               // K sub-offset for A/B fragments

    // --- Hoist B fragments (W columns) into registers: 2 N-tiles x 5 K-groups
    v2f bf[2][5];
#pragma unroll
    for (int nt = 0; nt < 2; ++nt) {
        const int n = wave * 32 + nt * 16 + m16;
#pragma unroll
        for (int kg = 0; kg < 5; ++kg) {
            const int k = kg * 4 + k0;
            v2f t;
            t.x = (k     < KDIM) ? W[(size_t)k       * HIDDEN + n] : 0.0f;
            t.y = (k + 1 < KDIM) ? W[(size_t)(k + 1) * HIDDEN + n] : 0.0f;
            bf[nt][kg] = t;
        }
    }
    // Bias pre-scaled by TANH_C so the whole tanh prologue is one FMA.
    const float biasC0 = bias[wave * 32 + m16]      * TANH_C;
    const float biasC1 = bias[wave * 32 + 16 + m16] * TANH_C;

    // --- Per-thread scan state: thread tid == hidden unit h
    const int h = tid;
    const float lam = fast_sigmoid(lam_p[h]);
    const float oml = 1.0f - lam;
    float hstate = PHASE2 ? hinit[((size_t)b * NCHUNK + c) * HIDDEN + h] : 0.0f;

    const int t0 = c * CHUNK;
    const float* ub = u + (size_t)b * L_SZ * U_DIM;
    const float* yb = y + (size_t)b * L_SZ * Y_DIM;

    for (int tile = 0; tile < NTILES; ++tile) {
        const int tbase = t0 + tile * TILE;

        // Cooperative stage of 16x17 input tile into LDS (zero-padded to 20)
        for (int i = tid; i < TILE * KPAD; i += 128) {
            const int m = i / KPAD;
            const int d = i % KPAD;
            const int t = tbase + m;
            float v = 0.0f;
            if (d < U_DIM)     v = ub[(size_t)t * U_DIM + d];
            else if (d < KDIM) v = yb[(size_t)t * Y_DIM + (d - U_DIM)];
            s_inp[m][d] = v;
        }
        __syncthreads();

        // A fragments: 16x4 fp32 per K-group (lane m16 = timestep row)
        v2f af[5];
#pragma unroll
        for (int kg = 0; kg < 5; ++kg) {
            v2f a;
            a.x = s_inp[m16][kg * 4 + k0];
            a.y = s_inp[m16][kg * 4 + k0 + 1];
            af[kg] = a;
        }

        // Two 16x16 output tiles per wave: D = sum_kg A_kg x B_kg (exact fp32)
#pragma unroll
        for (int nt = 0; nt < 2; ++nt) {
            v8f acc = {};
#pragma unroll
            for (int kg = 0; kg < 5; ++kg) {
                acc = __builtin_amdgcn_wmma_f32_16x16x4_f32(
                    /*neg_a=*/false, af[kg],
                    /*neg_b=*/false, bf[nt][kg],
                    /*c_mod=*/(short)0, acc,
                    /*reuse_a=*/false, /*reuse_b=*/false);
            }
            const float bnC = nt ? biasC1 : biasC0;
            const int   n   = wave * 32 + nt * 16 + m16;
#pragma unroll
            for (int v = 0; v < 8; ++v) {
                const int M = v + 8 * hi;  // C/D layout: lanes 16..31 hold M+8
                s_z[M][n] = fused_bias_tanh(acc[v], bnC);
            }
        }
        __syncthreads();

        // Leaky-integrator scan: thread h walks 16 timesteps in order
#pragma unroll
        for (int m = 0; m < TILE; ++m) {
            const float zv = s_z[m][h];
            hstate = lam * hstate + oml * zv;
            if (PHASE2) {
                // Streaming write-once output: keep it out of the way of L2.
                __builtin_nontemporal_store(
                    hstate, &h_seq[((size_t)b * L_SZ + (tbase + m)) * HIDDEN + h]);
            }
        }
        __syncthreads();  // protect LDS reuse next tile
    }

    if (PHASE2) {
        if (c == NCHUNK - 1) h_last[(size_t)b * HIDDEN + h] = hstate;
    } else {
        local_out[((size_t)b * NCHUNK + c) * HIDDEN + h] = hstate;
    }
}

// ---------------------------------------------------------------------------
// Sequential combine across chunks (exact): h_init[c] = lam^T * h_init[c-1]
// + local_end[c-1]. Only NCHUNK=32 steps per (b,h) pair.
// ---------------------------------------------------------------------------
__global__ __launch_bounds__(128)
void hydro_combine_kernel(const float* __restrict__ h0,
                          const float* __restrict__ lam_p,
                          const float* __restrict__ local,   // (B, NCHUNK, H)
                          float* __restrict__ hinit)         // (B, NCHUNK, H)
{
    const int b = blockIdx.x;
    const int h = threadIdx.x;
    const float lam  = fast_sigmoid(lam_p[h]);
    const float lamT = powf(lam, (float)CHUNK);
    float hcur = h0[(size_t)b * HIDDEN + h];
    for (int c = 0; c < NCHUNK; ++c) {
        const size_t idx = ((size_t)b * NCHUNK + c) * HIDDEN + h;
        hinit[idx] = hcur;
        hcur = lamT * hcur + local[idx];
    }
}

extern "C" void kernel_launch(void* const* d_in, const int* in_sizes, int n_in,
                              void* d_out, int out_size, void* d_ws, size_t ws_size,
                              hipStream_t stream) {
    // setup_inputs order: u_eff_seq, y_seq, h0, W, b, lam_param
    const float* u    = (const float*)d_in[0];
    const float* y    = (const float*)d_in[1];
    const float* h0   = (const float*)d_in[2];
    const float* W    = (const float*)d_in[3];
    const float* bias = (const float*)d_in[4];
    const float* lamp = (const float*)d_in[5];

    float* h_seq  = (float*)d_out;                                // (B, L, H)
    float* h_last = h_seq + (size_t)B_SZ * L_SZ * HIDDEN;         // (B, H)

    float* local = (float*)d_ws;                                  // (B, NCHUNK, H)
    float* hinit = local + (size_t)B_SZ * NCHUNK * HIDDEN;        // (B, NCHUNK, H)

    dim3 grid(NCHUNK, B_SZ);   // 32 x 64 = 2048 blocks
    dim3 block(128);

    hydro_chunk_kernel<false><<<grid, block, 0, stream>>>(
        u, y, W, bias, lamp, nullptr, local, nullptr, nullptr);

    hydro_combine_kernel<<<B_SZ, 128, 0, stream>>>(h0, lamp, local, hinit);

    hydro_chunk_kernel<true><<<grid, block, 0, stream>>>(
        u, y, W, bias, lamp, hinit, nullptr, h_seq, h_last);
}